// TransformedReLU_70583492543072
// MI455X (gfx1250) — compile-verified
//
#include <hip/hip_runtime.h>
#include <stddef.h>
#include <stdint.h>

typedef __attribute__((ext_vector_type(2))) float v2f;
typedef __attribute__((ext_vector_type(4))) float v4f;
typedef __attribute__((ext_vector_type(8))) float v8f;
typedef __attribute__((ext_vector_type(4))) unsigned int v4u;
typedef __attribute__((ext_vector_type(4))) int v4i;
typedef __attribute__((ext_vector_type(8))) int v8i;

#define N_COLS   8192
#define E_ROWS   2048
#define E1_ROWS  2049   // 1 center + 2048 error rows
#define NROW4    (N_COLS / 4)

#define AS1 __attribute__((address_space(1)))
#define AS3 __attribute__((address_space(3)))

#if defined(__has_builtin)
#if __has_builtin(__builtin_amdgcn_tensor_load_to_lds) && \
    __has_builtin(__builtin_amdgcn_s_wait_tensorcnt)
#define HAVE_TDM 1
#warning "CDNA5 path: TDM tensor_load_to_lds"
#else
#define HAVE_TDM 0
#endif
#if __has_builtin(__builtin_amdgcn_global_load_async_to_lds_b64)
#define HAVE_ASYNC_LDS 1
#warning "CDNA5 path: global_load_async_to_lds available"
#else
#define HAVE_ASYNC_LDS 0
#endif
#else
#define HAVE_TDM 0
#define HAVE_ASYNC_LDS 0
#endif

// ---------------------------------------------------------------------------
// Kernel A: zero the worst-case padded "extra" block (N x N floats = 256 MB).
// Write-only and larger than the 192 MB L2 -> non-temporal b128 stores.
// ---------------------------------------------------------------------------
__global__ void k_zero_extra(v4f* __restrict__ dst) {
  size_t i = (size_t)blockIdx.x * blockDim.x + threadIdx.x;
  v4f z = (v4f)0.0f;
  __builtin_nontemporal_store(z, dst + i);
}

// ---------------------------------------------------------------------------
// Kernel B: partial column abs-sums via V_WMMA_F32_16X16X4_F32.
//   A = all-ones 16x4  =>  D[m][n] = sum_k B[k][n] + C[m][n]  (rows identical)
// 128 column-tiles (64 cols) x 8 row-chunks (256 rows) = 1024 waves.
//
// Staging: one TDM TENSOR_LOAD_TO_LDS per 4x64 tile (256 B/row contiguous,
// hardware-padded to an 80-dword LDS pitch via D# pad_interval=5/pad_amount=15
// so every ds_load_b32 B-fragment read is bank-conflict-free). Double
// buffered on TENSORcnt.  fabs applied when building B fragments.
// ---------------------------------------------------------------------------
#define WPB      4
#define PITCH    80
#define CHUNKS   8
#define CH_ROWS  (E_ROWS / CHUNKS)     // 256
#define STAGES   (CH_ROWS / 4)         // 64

__global__ void k_abssum_wmma(const float* __restrict__ xe,
                              float* __restrict__ partial) {
  const int lane = threadIdx.x & 31;
  const int wIB  = threadIdx.x >> 5;
  const int wave = blockIdx.x * WPB + wIB;        // 0..1023
  const int tile  = wave >> 3;
  const int chunk = wave & (CHUNKS - 1);
  const int col0 = tile * 64;
  const int l15  = lane & 15;
  const int half = lane >> 4;

  __shared__ float smem[WPB][2][4 * PITCH];

  v8f c[4];
  c[0] = (v8f)0.0f; c[1] = (v8f)0.0f; c[2] = (v8f)0.0f; c[3] = (v8f)0.0f;
  v2f a; a.x = 1.0f; a.y = 1.0f;

#if HAVE_TDM
  // ---- Tensor Data Mover path -------------------------------------------
  const uint64_t gbase =
      (uint64_t)(uintptr_t)(xe + (size_t)chunk * CH_ROWS * N_COLS + col0);
  uint32_t lds0 = (uint32_t)(uintptr_t)(AS3 char*)&smem[wIB][0][0];
  uint32_t lds1 = (uint32_t)(uintptr_t)(AS3 char*)&smem[wIB][1][0];
  lds0 = (uint32_t)__builtin_amdgcn_readfirstlane((int)lds0);
  lds1 = (uint32_t)__builtin_amdgcn_readfirstlane((int)lds1);

  // D# group1: data_size=4B, pad_enable, pad every 64 dwords by 16 dwords,
  // tensor_dim0=8192, tensor_dim1=2048, tile 64x4, dim0_stride=8192.
  const v8i g1 = { (int)((2u << 16) | (1u << 20) | (5u << 22) | (15u << 25)),
                   (int)(8192u << 16),   // tensor_dim0[15:0] -> bits 63:48
                   (int)(2048u << 16),   // dim0 hi(0) | tensor_dim1[15:0]
                   (int)(64u   << 16),   // dim1 hi(0) | tile_dim0 = 64
                   4,                    // tile_dim1 = 4, tile_dim2 = 0
                   8192,                 // tensor_dim0_stride[31:0]
                   0, 0 };
  const v4i gz = {0, 0, 0, 0};

  auto issue = [&](uint32_t ldsoff, int stage) {
    uint64_t ga = gbase + (uint64_t)stage * (4ull * N_COLS * 4ull);
    uint32_t lo = (uint32_t)__builtin_amdgcn_readfirstlane((int)(uint32_t)ga);
    uint32_t hi = (uint32_t)__builtin_amdgcn_readfirstlane((int)(uint32_t)(ga >> 32));
    v4u g0;
    g0.x = 1u;                               // count=1 (valid user descriptor)
    g0.y = ldsoff;                           // lds_addr
    g0.z = lo;                               // global_addr[31:0]
    g0.w = (hi & 0x1FFFFFFu) | (2u << 30);   // global_addr[56:32] | type=2
#if __clang_major__ >= 23
    const v8i z8 = {0, 0, 0, 0, 0, 0, 0, 0};
    __builtin_amdgcn_tensor_load_to_lds(g0, g1, gz, gz, z8, 0);
#else
    __builtin_amdgcn_tensor_load_to_lds(g0, g1, gz, gz, 0);
#endif
  };

  issue(lds0, 0);
  issue(lds1, 1);
  for (int k = 0; k < STAGES; ++k) {
    const int buf = k & 1;
    __builtin_amdgcn_s_wait_tensorcnt(1);    // oldest TDM op retired
    asm volatile("" ::: "memory");           // don't hoist the LDS reads
    const float* s = &smem[wIB][buf][0];
    #pragma unroll
    for (int t = 0; t < 4; ++t) {
      v2f b;
      b.x = __builtin_fabsf(s[(2 * half)     * PITCH + t * 16 + l15]);
      b.y = __builtin_fabsf(s[(2 * half + 1) * PITCH + t * 16 + l15]);
      c[t] = __builtin_amdgcn_wmma_f32_16x16x4_f32(false, a, false, b,
                                                   (short)0, c[t], false, false);
    }
    asm volatile("s_wait_dscnt 0x0" ::: "memory");  // reads retired pre-refill
    if (k + 2 < STAGES) issue(buf ? lds1 : lds0, k + 2);
  }
#elif HAVE_ASYNC_LDS
  // ---- async global->LDS path -------------------------------------------
  const float* gsrc = xe + (size_t)chunk * CH_ROWS * N_COLS + col0 + 2 * lane;
  float* lbase = &smem[wIB][0][2 * lane];
  auto issue = [&](int buf, int stage) {
    const float* g = gsrc + (size_t)stage * 4 * N_COLS;
    float* l = lbase + buf * (4 * PITCH);
    #pragma unroll
    for (int r = 0; r < 4; ++r)
      __builtin_amdgcn_global_load_async_to_lds_b64(
          (AS1 void*)(g + (size_t)r * N_COLS), (AS3 void*)(l + r * PITCH), 0, 0);
  };
  issue(0, 0);
  issue(1, 1);
  for (int k = 0; k < STAGES; ++k) {
    const int buf = k & 1;
    asm volatile("s_wait_asynccnt 0x4" ::: "memory");
    const float* s = &smem[wIB][buf][0];
    #pragma unroll
    for (int t = 0; t < 4; ++t) {
      v2f b;
      b.x = __builtin_fabsf(s[(2 * half)     * PITCH + t * 16 + l15]);
      b.y = __builtin_fabsf(s[(2 * half + 1) * PITCH + t * 16 + l15]);
      c[t] = __builtin_amdgcn_wmma_f32_16x16x4_f32(false, a, false, b,
                                                   (short)0, c[t], false, false);
    }
    asm volatile("s_wait_dscnt 0x0" ::: "memory");
    if (k + 2 < STAGES) issue(buf, k + 2);
  }
#else
  // ---- synchronous staging fallback -------------------------------------
  const float* gsrc = xe + (size_t)chunk * CH_ROWS * N_COLS + col0 + 2 * lane;
  float* lbase = &smem[wIB][0][2 * lane];
  for (int k = 0; k < STAGES; ++k) {
    const float* g = gsrc + (size_t)k * 4 * N_COLS;
    #pragma unroll
    for (int r = 0; r < 4; ++r)
      *(v2f*)(lbase + r * PITCH) = *(const v2f*)(g + (size_t)r * N_COLS);
    const float* s = &smem[wIB][0][0];
    #pragma unroll
    for (int t = 0; t < 4; ++t) {
      v2f b;
      b.x = __builtin_fabsf(s[(2 * half)     * PITCH + t * 16 + l15]);
      b.y = __builtin_fabsf(s[(2 * half + 1) * PITCH + t * 16 + l15]);
      c[t] = __builtin_amdgcn_wmma_f32_16x16x4_f32(false, a, false, b,
                                                   (short)0, c[t], false, false);
    }
  }
#endif

  if (lane < 16) {
    #pragma unroll
    for (int t = 0; t < 4; ++t)
      partial[(size_t)chunk * N_COLS + col0 + t * 16 + lane] = c[t][0];
  }
}

// ---------------------------------------------------------------------------
// Kernel C: fold 8 partials -> abs_sum; write center row; block scan of the
// crossing mask; scatter -lam*lower/2 at row (E1 + exclusive_prefix).
// ---------------------------------------------------------------------------
__global__ void k_center_scan_scatter(const float* __restrict__ x0,
                                      const float* __restrict__ lam,
                                      const float* __restrict__ partial,
                                      float* __restrict__ out) {
  const int t = threadIdx.x;            // 0..1023
  const int base = t * 8;
  __shared__ int psum[1024];

  int   has[8];
  float val[8];
  int lsum = 0;
  #pragma unroll
  for (int i = 0; i < 8; ++i) {
    const int n = base + i;
    float a = 0.0f;
    #pragma unroll
    for (int j = 0; j < CHUNKS; ++j) a += partial[(size_t)j * N_COLS + n];
    const float c0 = x0[n];
    const float l  = lam[n];
    const float lower = c0 - a;
    const float upper = c0 + a;
    const int   h = (lower < 0.0f) && (upper > 0.0f);
    has[i] = h;
    val[i] = -l * lower * 0.5f;
    lsum  += h;
    out[n] = l * c0 - l * lower * 0.5f;
  }
  psum[t] = lsum;
  __syncthreads();

  for (int off = 1; off < 1024; off <<= 1) {
    int v   = psum[t];
    int add = (t >= off) ? psum[t - off] : 0;
    __syncthreads();
    psum[t] = v + add;
    __syncthreads();
  }
  int run = (t == 0) ? 0 : psum[t - 1];

  #pragma unroll
  for (int i = 0; i < 8; ++i) {
    if (has[i]) {
      out[(size_t)(E1_ROWS + run) * N_COLS + (base + i)] = val[i];
      ++run;
    }
  }
}

// ---------------------------------------------------------------------------
// Kernel D: streaming scale of the 2048 error rows (NT b128 + prefetch).
// ---------------------------------------------------------------------------
__global__ void k_scale_errs(const v4f* __restrict__ xe,
                             const v4f* __restrict__ lam4,
                             v4f* __restrict__ oute) {
  size_t i = (size_t)blockIdx.x * blockDim.x + threadIdx.x;
  __builtin_prefetch((const void*)(xe + i + 8192), 0, 1);
  v4f l = lam4[i & (NROW4 - 1)];
  v4f v = __builtin_nontemporal_load(xe + i);
  __builtin_nontemporal_store(v * l, oute + i);
}

extern "C" void kernel_launch(void* const* d_in, const int* in_sizes, int n_in,
                              void* d_out, int out_size, void* d_ws, size_t ws_size,
                              hipStream_t stream) {
  (void)in_sizes; (void)n_in; (void)out_size; (void)ws_size;

  const float* x   = (const float*)d_in[0];   // [1, 2049, 8192]
  const float* lam = (const float*)d_in[1];   // [1, 8192]
  float*       out = (float*)d_out;           // [1, 10241, 8192]
  float*       ws  = (float*)d_ws;            // 8 x 8192 floats = 256 KB

  const float* x_err   = x   + N_COLS;
  float*       out_err = out + N_COLS;
  v4f*         extra4  = (v4f*)(out + (size_t)E1_ROWS * N_COLS);

  k_zero_extra<<<65536, 256, 0, stream>>>(extra4);
  k_abssum_wmma<<<256, 128, 0, stream>>>(x_err, ws);
  k_center_scan_scatter<<<1, 1024, 0, stream>>>(x, lam, ws, out);
  k_scale_errs<<<16384, 256, 0, stream>>>((const v4f*)x_err, (const v4f*)lam,
                                          (v4f*)out_err);
}